// QuantumQuanvolutionFilter_65481071398039
// MI455X (gfx1250) — compile-verified
//
#include <hip/hip_runtime.h>
#include <math.h>

typedef __attribute__((ext_vector_type(2))) float v2f;
typedef __attribute__((ext_vector_type(8))) float v8f;

// ---------------------------------------------------------------------------
// Kernel 1: build the fixed 16x16 complex circuit matrix U = G8*...*G1 from
// theta[8]. Thread t propagates identity column e_t through the gate list.
// Output layout in d_ws: U[0..255] = Re (row-major U[z][col]), U[256..511]=Im.
// ---------------------------------------------------------------------------
__global__ void quanv_prep_kernel(const float* __restrict__ theta,
                                  float* __restrict__ U) {
  const int t = threadIdx.x;
  if (t >= 16) return;

  float vr[16], vi[16];
#pragma unroll
  for (int z = 0; z < 16; ++z) { vr[z] = (z == t) ? 1.f : 0.f; vi[z] = 0.f; }

  // op list: (ry,0)(rx,1)(rz,2)(crx,0->1)(ry,3)(cry,2->3)(rx,0)(crz,1->2)
  const int kind[8] = {0, 1, 2, 1, 0, 0, 1, 2};   // 0=ry 1=rx 2=rz
  const int targ[8] = {0, 1, 2, 1, 3, 3, 0, 2};
  const int ctrl[8] = {-1, -1, -1, 0, -1, 2, -1, 1};

#pragma unroll
  for (int g = 0; g < 8; ++g) {
    const float th = 0.5f * theta[g];
    const float c = cosf(th), s = sinf(th);
    float g00r, g00i, g01r, g01i, g10r, g10i, g11r, g11i;
    if (kind[g] == 0) {          // RY: [[c,-s],[s,c]]
      g00r = c;  g00i = 0.f; g01r = -s;  g01i = 0.f;
      g10r = s;  g10i = 0.f; g11r = c;   g11i = 0.f;
    } else if (kind[g] == 1) {   // RX: [[c,-i s],[-i s,c]]
      g00r = c;  g00i = 0.f; g01r = 0.f; g01i = -s;
      g10r = 0.f; g10i = -s; g11r = c;   g11i = 0.f;
    } else {                     // RZ: [[c-is,0],[0,c+is]]
      g00r = c;  g00i = -s;  g01r = 0.f; g01i = 0.f;
      g10r = 0.f; g10i = 0.f; g11r = c;  g11i = s;
    }
    const int tm = 8 >> targ[g];                       // wire w -> bit (3-w)
    const int cm = (ctrl[g] >= 0) ? (8 >> ctrl[g]) : 0;
#pragma unroll
    for (int z = 0; z < 16; ++z) {
      if (z & tm) continue;                 // visit target-bit==0 element
      if (cm && !(z & cm)) continue;        // controlled: need control bit==1
      const int z1 = z | tm;
      const float v0r = vr[z], v0i = vi[z], v1r = vr[z1], v1i = vi[z1];
      vr[z]  = g00r * v0r - g00i * v0i + g01r * v1r - g01i * v1i;
      vi[z]  = g00r * v0i + g00i * v0r + g01r * v1i + g01i * v1r;
      vr[z1] = g10r * v0r - g10i * v0i + g11r * v1r - g11i * v1i;
      vi[z1] = g10r * v0i + g10i * v0r + g11r * v1i + g11i * v1r;
    }
  }
#pragma unroll
  for (int z = 0; z < 16; ++z) {
    U[z * 16 + t]       = vr[z];   // Re
    U[256 + z * 16 + t] = vi[z];   // Im
  }
}

// ---------------------------------------------------------------------------
// Kernel 2: per wave, one tile of 16 patches.
//   y_re = U_re * S, y_im = U_im * S   via 4 chained V_WMMA_F32_16X16X4_F32
//   meas[w] = sum_z sign(w,z) * (y_re^2 + y_im^2)
// A-fragment layout (ISA 16x4 f32 A): lanes 0-15 row M=lane, VGPR0=K0,VGPR1=K1;
// lanes 16-31 VGPR0=K2, VGPR1=K3. B (4x16) mirrors with N per lane.
// C/D 16x16 f32: VGPR v holds [M=v][N=lane] (lanes<16) / [M=v+8][N=lane-16].
// ---------------------------------------------------------------------------
__global__ __launch_bounds__(256) void quanv_main_kernel(
    const float* __restrict__ x, const float* __restrict__ U,
    float* __restrict__ out, int nPatches, int nTiles) {
  __shared__ float S[8][16][16];   // [wave][z][patch-in-tile], 8 KB

  const int lane = threadIdx.x & 31;
  const int wave = threadIdx.x >> 5;
  const int n    = lane & 15;   // patch-in-tile == column == row index for A
  const int hi   = lane >> 4;   // half-wave select (K/M split)

  int tile = blockIdx.x * 8 + wave;
  if (tile >= nTiles) tile = nTiles - 1;        // keep all waves at barrier
  const int p     = tile * 16 + n;
  const bool valid = (p < nPatches);
  const int pc    = valid ? p : (nPatches - 1); // branchless clamp

  // patch pc -> image b, patch row i, patch col j; angles are the 2x2 pixels
  const int b = pc / 196;
  const int r = pc - b * 196;
  const int i = r / 14;
  const int j = r - i * 14;
  const float* xp = x + (size_t)b * 784 + (size_t)(2 * i) * 28 + (size_t)(2 * j);
  const float a0 = xp[0], a1 = xp[1], a2 = xp[28], a3 = xp[29];

  float c0, s0, c1, s1, c2, s2, c3, s3;
  sincosf(0.5f * a0, &s0, &c0);
  sincosf(0.5f * a1, &s1, &c1);
  sincosf(0.5f * a2, &s2, &c2);
  sincosf(0.5f * a3, &s3, &c3);

  // product state amplitudes: wire w bit of z is (z >> (3-w)) & 1
  // each half-wave fills its 8 rows; writes are 16 consecutive words -> no
  // bank conflicts
#pragma unroll
  for (int zz = 0; zz < 8; ++zz) {
    const int z = hi * 8 + zz;
    const float amp = ((z & 8) ? s0 : c0) * ((z & 4) ? s1 : c1) *
                      ((z & 2) ? s2 : c2) * ((z & 1) ? s3 : c3);
    S[wave][z][n] = amp;
  }
  __syncthreads();

  v8f accR = {0.f, 0.f, 0.f, 0.f, 0.f, 0.f, 0.f, 0.f};
  v8f accI = {0.f, 0.f, 0.f, 0.f, 0.f, 0.f, 0.f, 0.f};
#pragma unroll
  for (int k = 0; k < 4; ++k) {
    const int kk = 4 * k + 2 * hi;
    v2f aR, aI, bv;
    aR.x = U[n * 16 + kk];           aR.y = U[n * 16 + kk + 1];
    aI.x = U[256 + n * 16 + kk];     aI.y = U[256 + n * 16 + kk + 1];
    bv.x = S[wave][kk][n];           bv.y = S[wave][kk + 1][n];
    accR = __builtin_amdgcn_wmma_f32_16x16x4_f32(
        false, aR, false, bv, (short)0, accR, false, false);
    accI = __builtin_amdgcn_wmma_f32_16x16x4_f32(
        false, aI, false, bv, (short)0, accI, false, false);
  }

  // probs + signed reduction; this lane holds rows z = v + 8*hi, column n.
  // bit3(z) == hi (uniform per half-wave); bits 2..0 come from v (compile
  // time per unrolled iteration).
  float sumAll = 0.f, e1 = 0.f, e2 = 0.f, e3 = 0.f;
#pragma unroll
  for (int v = 0; v < 8; ++v) {
    const float yr = accR[v], yi = accI[v];
    const float pr = yr * yr + yi * yi;
    sumAll += pr;
    e1 += (v & 4) ? -pr : pr;
    e2 += (v & 2) ? -pr : pr;
    e3 += (v & 1) ? -pr : pr;
  }
  float e0 = hi ? -sumAll : sumAll;

  // combine the two half-wave partial sums (rows 0-7 with rows 8-15)
  e0 += __shfl_xor(e0, 16);
  e1 += __shfl_xor(e1, 16);
  e2 += __shfl_xor(e2, 16);
  e3 += __shfl_xor(e3, 16);

  if (hi == 0 && valid) {
    float4 o = make_float4(e0, e1, e2, e3);
    *(float4*)(out + (size_t)p * 4) = o;   // 16B aligned, contiguous per tile
  }
}

// ---------------------------------------------------------------------------
extern "C" void kernel_launch(void* const* d_in, const int* in_sizes, int n_in,
                              void* d_out, int out_size, void* d_ws,
                              size_t ws_size, hipStream_t stream) {
  const float* x     = (const float*)d_in[0];   // [B,28,28] f32
  const float* theta = (const float*)d_in[1];   // [8] f32
  float* out = (float*)d_out;                   // [B,784] f32
  float* U   = (float*)d_ws;                    // 512 floats (Re|Im)

  const int B        = in_sizes[0] / 784;
  const int nPatches = B * 196;
  const int nTiles   = (nPatches + 15) / 16;
  const int nBlocks  = (nTiles + 7) / 8;

  quanv_prep_kernel<<<1, 32, 0, stream>>>(theta, U);
  quanv_main_kernel<<<nBlocks, 256, 0, stream>>>(x, U, out, nPatches, nTiles);
}